// FunctionCorrelationTranspose_40209483825164
// MI455X (gfx1250) — compile-verified
//
#include <hip/hip_runtime.h>

typedef __attribute__((ext_vector_type(2))) float v2f;
typedef __attribute__((ext_vector_type(8))) float v8f;

#define MD   4
#define HGT  128
#define WID  128
#define CCH  256
#define CG   64          // channels per workgroup
#define KDIM 144         // 12x12 second patch -> GEMM K
#define STR  148         // padded LDS row stride (floats); 148%64=20 -> conflict-free b64 frag loads

__global__ __launch_bounds__(128) void corr_transpose_wmma(
    const float* __restrict__ inp,      // (B, 81, H, W)
    const float* __restrict__ sec,      // (B, 256, H, W)
    float* __restrict__ out)            // (B, 256, H, W)
{
    __shared__ float sA[CG * STR];      // second patch, [channel][k]
    __shared__ float sB[16 * STR];      // weights^T,    [pixel n][k]

    const int tid  = threadIdx.x;
    const int tile = blockIdx.x;        // 0..1023  (32x32 tiles of 4x4 pixels)
    const int cg   = blockIdx.y;        // 0..3     (channel group of 64)
    const int bb   = blockIdx.z;        // 0..7     (batch)
    const int h0   = (tile >> 5) << 2;
    const int w0   = (tile & 31) << 2;

    // ---- stage A: 12x12 zero-padded patch of `second` for 64 channels ----
    {
        const float* s2 = sec + ((size_t)bb * CCH + (size_t)cg * CG) * (HGT * WID);
        for (int i = tid; i < CG * KDIM; i += 128) {
            int c  = i / KDIM;
            int k  = i - c * KDIM;
            int hy = k / 12;
            int hx = k - hy * 12;
            int gh = h0 - MD + hy;
            int gw = w0 - MD + hx;
            float v = 0.0f;
            if (gh >= 0 && gh < HGT && gw >= 0 && gw < WID)
                v = s2[(size_t)c * (HGT * WID) + gh * WID + gw];
            sA[c * STR + k] = v;
        }
    }
    // ---- stage B: zero, then scatter the 81 input taps per pixel ----
    for (int i = tid; i < 16 * STR; i += 128) sB[i] = 0.0f;
    __syncthreads();
    {
        const float* ip = inp + (size_t)bb * 81 * (HGT * WID);
        for (int i = tid; i < 81 * 16; i += 128) {
            int d  = i >> 4;            // tap index 0..80
            int n  = i & 15;            // pixel in 4x4 tile
            int ty = n >> 2, tx = n & 3;
            int dy = d / 9;
            int dx = d - dy * 9;
            int k  = (dy + ty) * 12 + (dx + tx);   // position in 12x12 patch
            float v = ip[(size_t)d * (HGT * WID) + (h0 + ty) * WID + (w0 + tx)];
            sB[n * STR + k] = v;
        }
    }
    __syncthreads();

    // ---- per-wave 16(channels) x 16(pixels) tile, K=144 via 36x WMMA f32 16x16x4 ----
    const int wave  = tid >> 5;
    const int lane  = tid & 31;
    const int row   = lane & 15;        // M for A-frag, N for B/D-frags
    const int khalf = lane >> 4;        // K pair select within each K=4 step

    const float* aptr = &sA[(wave * 16 + row) * STR];
    const float* bptr = &sB[row * STR];

    v8f acc = {0.f, 0.f, 0.f, 0.f, 0.f, 0.f, 0.f, 0.f};
    #pragma unroll
    for (int k0 = 0; k0 < KDIM; k0 += 4) {
        const int kk = k0 + 2 * khalf;
        v2f a = *(const v2f*)(aptr + kk);   // ds_load_b64, conflict-free
        v2f b = *(const v2f*)(bptr + kk);
        acc = __builtin_amdgcn_wmma_f32_16x16x4_f32(
            /*neg_a=*/false, a, /*neg_b=*/false, b,
            /*c_mod=*/(short)0, acc, /*reuse_a=*/false, /*reuse_b=*/false);
    }

    // ---- store D (scaled by 1/C): VGPR v holds M = v + 8*khalf, N = lane&15 ----
    const int ty = row >> 2, tx = row & 3;
    float* op = out + (((size_t)bb * CCH + cg * CG + wave * 16 + 8 * khalf) * (size_t)(HGT * WID))
                    + (h0 + ty) * WID + (w0 + tx);
    #pragma unroll
    for (int v = 0; v < 8; ++v)
        op[(size_t)v * (HGT * WID)] = acc[v] * (1.0f / 256.0f);
}

extern "C" void kernel_launch(void* const* d_in, const int* in_sizes, int n_in,
                              void* d_out, int out_size, void* d_ws, size_t ws_size,
                              hipStream_t stream) {
    (void)in_sizes; (void)n_in; (void)d_ws; (void)ws_size; (void)out_size;
    const float* inp = (const float*)d_in[0];   // (8, 81, 128, 128) f32
    const float* sec = (const float*)d_in[1];   // (8, 256, 128, 128) f32
    float*       out = (float*)d_out;           // (8, 256, 128, 128) f32

    dim3 grid(1024, 4, 8);   // 4x4-pixel tiles  x  channel groups  x  batch
    corr_transpose_wmma<<<grid, 128, 0, stream>>>(inp, sec, out);
}